// GNNEncoder_30124900614621
// MI455X (gfx1250) — compile-verified
//
#include <hip/hip_runtime.h>

typedef float v2f __attribute__((ext_vector_type(2)));
typedef float v8f __attribute__((ext_vector_type(8)));

#define IN_CH 256
#define HID   128

// ---------------------------------------------------------------- utilities
__global__ void gcn_fill_ones(float* __restrict__ p, int n) {
    int i = blockIdx.x * blockDim.x + threadIdx.x;
    if (i < n) p[i] = 1.0f;
}

__global__ void gcn_degree_accum(const int* __restrict__ dst, int E,
                                 float* __restrict__ deg) {
    int e = blockIdx.x * blockDim.x + threadIdx.x;
    if (e < E) atomicAdd(&deg[dst[e]], 1.0f);   // global_atomic_add_f32, L2-resident
}

__global__ void gcn_rsqrt_inplace(float* __restrict__ d, int n) {
    int i = blockIdx.x * blockDim.x + threadIdx.x;
    if (i < n) d[i] = rsqrtf(d[i]);             // deg >= 1, safe
}

// ------------------------------------------------- fp32 WMMA GEMM (h = A @ W)
// C[M x HID] = A[M x K] @ W[K x HID], all row-major fp32.
// One wave computes a 16x16 tile via V_WMMA_F32_16X16X4_F32 (K step 4).
// A frag (16x4): lanes 0-15 -> K=0,1 ; lanes 16-31 -> K=2,3 (2 VGPRs)
// B frag (4x16): lanes 0-15 -> K=0,1 ; lanes 16-31 -> K=2,3 (2 VGPRs)
// C/D (16x16):  VGPR r: lanes 0-15 = row r, lanes 16-31 = row r+8
template <int K>
__global__ void gcn_wmma_gemm(const float* __restrict__ A,
                              const float* __restrict__ W,
                              float* __restrict__ C, int M) {
    const int lane = threadIdx.x & 31;
    const int wave = threadIdx.x >> 5;
    const int half = lane >> 4;          // 0: lanes 0-15, 1: lanes 16-31
    const int l    = lane & 15;
    const int m0   = blockIdx.x * 16;
    const int n0   = (blockIdx.y * (blockDim.x >> 5) + wave) * 16;

    int arow = m0 + l;
    if (arow >= M) arow = M - 1;         // clamp (keeps EXEC all-ones for WMMA)
    const float* __restrict__ arow_p = A + (size_t)arow * K;

    v8f acc = {};
#pragma unroll 8
    for (int k = 0; k < K; k += 4) {
        const int ka = k + half * 2;
        v2f a, b;
        a.x = arow_p[ka];
        a.y = arow_p[ka + 1];
        b.x = W[(size_t)ka * HID + n0 + l];
        b.y = W[(size_t)(ka + 1) * HID + n0 + l];
        acc = __builtin_amdgcn_wmma_f32_16x16x4_f32(
            /*neg_a=*/false, a, /*neg_b=*/false, b,
            /*c_mod=*/(short)0, acc, /*reuse_a=*/false, /*reuse_b=*/false);
    }

#pragma unroll
    for (int r = 0; r < 8; ++r) {
        const int row = m0 + r + half * 8;
        if (row < M) C[(size_t)row * HID + n0 + l] = acc[r];
    }
}

// -------------------------------------------- edge scatter: agg[dst] += h[src]*norm
// One wave32 per edge; each lane carries a float4 (32*4 = 128 features).
// h and agg are ~25.6 MB each: resident in the 192 MB L2, so gathers and the
// atomic reductions run at L2 bandwidth, not HBM.
__global__ void gcn_scatter_edges(const int* __restrict__ src,
                                  const int* __restrict__ dst, int E,
                                  const float* __restrict__ h,
                                  const float* __restrict__ dinv,
                                  float* __restrict__ agg) {
    const int widx = (blockIdx.x * blockDim.x + threadIdx.x) >> 5;  // edge id
    const int lane = threadIdx.x & 31;
    if (widx >= E) return;
    const int s = src[widx];
    const int d = dst[widx];
    const float norm = dinv[s] * dinv[d];

    const float4 v = ((const float4*)(h + (size_t)s * HID))[lane];
    float* a = agg + (size_t)d * HID + lane * 4;
    atomicAdd(a + 0, v.x * norm);
    atomicAdd(a + 1, v.y * norm);
    atomicAdd(a + 2, v.z * norm);
    atomicAdd(a + 3, v.w * norm);
}

// -------------------- finalize: out = relu(agg + h*dinv^2 + b)  (elementwise)
__global__ void gcn_finalize(const float* __restrict__ agg,
                             const float* __restrict__ h,
                             const float* __restrict__ dinv,
                             const float* __restrict__ b,
                             float* __restrict__ out, long total) {
    long i = (long)blockIdx.x * blockDim.x + threadIdx.x;
    if (i >= total) return;
    const int node = (int)(i >> 7);      // HID == 128
    const int f    = (int)(i & 127);
    const float di = dinv[node];
    const float v  = agg[i] + h[i] * di * di + b[f];
    out[i] = fmaxf(v, 0.0f);
}

// ---------------------------------------------------------------- launcher
extern "C" void kernel_launch(void* const* d_in, const int* in_sizes, int n_in,
                              void* d_out, int out_size, void* d_ws, size_t ws_size,
                              hipStream_t stream) {
    const float* x  = (const float*)d_in[0];
    const int*   ei = (const int*)d_in[1];
    const float* W1 = (const float*)d_in[2];
    const float* b1 = (const float*)d_in[3];
    const float* W2 = (const float*)d_in[4];
    const float* b2 = (const float*)d_in[5];
    float* out = (float*)d_out;

    const int N = in_sizes[0] / IN_CH;     // 50000
    const int E = in_sizes[1] / 2;         // 1.6M
    const int* src = ei;
    const int* dst = ei + E;

    char* ws = (char*)d_ws;
    float* dinv = (float*)ws;
    const size_t off    = (((size_t)N * 4) + 255) & ~(size_t)255;
    const size_t hbytes = (size_t)N * HID * sizeof(float);
    float* hbuf   = (float*)(ws + off);            // GEMM output / layer-2 GEMM output
    float* aggbuf = (float*)(ws + off + hbytes);   // scatter accumulator / relu buffer

    const int T = 256;
    const long totNH = (long)N * HID;

    // --- symmetric normalization: dinv = rsqrt(deg + 1)
    gcn_fill_ones<<<(N + T - 1) / T, T, 0, stream>>>(dinv, N);
    gcn_degree_accum<<<(E + T - 1) / T, T, 0, stream>>>(dst, E, dinv);
    gcn_rsqrt_inplace<<<(N + T - 1) / T, T, 0, stream>>>(dinv, N);

    const dim3 gemm_grid((N + 15) / 16, HID / 16 / 4);   // 4 waves/block x 2 blocks in y cover 128 cols
    const int  gemm_block = 128;
    const long scat_threads = (long)E * 32;
    const int  scat_blocks  = (int)((scat_threads + T - 1) / T);

    // --- layer 1
    gcn_wmma_gemm<IN_CH><<<gemm_grid, gemm_block, 0, stream>>>(x, W1, hbuf, N);
    hipMemsetAsync(aggbuf, 0, hbytes, stream);
    gcn_scatter_edges<<<scat_blocks, T, 0, stream>>>(src, dst, E, hbuf, dinv, aggbuf);
    // relu1 written in-place into aggbuf (it becomes layer-2 input)
    gcn_finalize<<<(int)((totNH + T - 1) / T), T, 0, stream>>>(aggbuf, hbuf, dinv, b1,
                                                              aggbuf, totNH);
    // --- layer 2
    gcn_wmma_gemm<HID><<<gemm_grid, gemm_block, 0, stream>>>(aggbuf, W2, hbuf, N);
    hipMemsetAsync(aggbuf, 0, hbytes, stream);
    gcn_scatter_edges<<<scat_blocks, T, 0, stream>>>(src, dst, E, hbuf, dinv, aggbuf);
    gcn_finalize<<<(int)((totNH + T - 1) / T), T, 0, stream>>>(aggbuf, hbuf, dinv, b2,
                                                              out, totNH);
}